// GIN_35914516529300
// MI455X (gfx1250) — compile-verified
//
#include <hip/hip_runtime.h>
#include <math.h>

// ---------------------------------------------------------------------------
// GIN 2-layer forward for MI455X (gfx1250).
// fp32 WMMA (v_wmma_f32_16x16x4_f32) for both GEMMs; BN folded into weights.
// ---------------------------------------------------------------------------

typedef float v2f __attribute__((ext_vector_type(2)));
typedef float v8f __attribute__((ext_vector_type(8)));

#define GN 100000
#define GE 1600000
#define GD 128
#define GH 32
#define BN_EPS 1e-5f
#define NEG_SLOPE 0.01f

// ---------------- elementwise helpers ----------------

__global__ void copy_f32(float* __restrict__ dst, const float* __restrict__ src, int n) {
    int i = blockIdx.x * blockDim.x + threadIdx.x;
    if (i < n) dst[i] = src[i];
}

__global__ void zero_f32(float* __restrict__ p, int n) {
    int i = blockIdx.x * blockDim.x + threadIdx.x;
    if (i < n) p[i] = 0.0f;
}

// ---------------- edge scatter: agg[dst] += x[src] ----------------
// one thread per (edge, 4-float chunk): E*32 threads total.

__global__ void __launch_bounds__(256)
scatter_edges(float* __restrict__ agg, const float* __restrict__ xsrc,
              const int* __restrict__ src, const int* __restrict__ dst, int total) {
    int t = blockIdx.x * blockDim.x + threadIdx.x;
    if (t >= total) return;
    int e = t >> 5;
    int c = (t & 31) << 2;
    int s = src[e];
    int d = dst[e];
    const float4 v = *(const float4*)(xsrc + (size_t)s * GD + c);
    float* p = agg + (size_t)d * GD + c;
    atomicAdd(p + 0, v.x);
    atomicAdd(p + 1, v.y);
    atomicAdd(p + 2, v.z);
    atomicAdd(p + 3, v.w);
}

// ---------------- GEMM1: H = ReLU((x+agg) @ W1 + b1), + BN stats ----------------
// one wave per 16-row tile; K=128 in steps of 4; 16x32 output per wave.

__global__ void __launch_bounds__(256)
gemm1_relu_stats(const float* __restrict__ A, const float* __restrict__ W1,
                 const float* __restrict__ b1, float* __restrict__ Hout,
                 float* __restrict__ stats, int ntiles) {
    int wave = (blockIdx.x << 3) + (threadIdx.x >> 5);
    if (wave >= ntiles) return;          // wave-uniform exit (EXEC stays all-1s)
    int lane = threadIdx.x & 31;
    int half = lane >> 4;
    int l16  = lane & 15;

    const float* arow = A + (size_t)(wave * 16 + l16) * GD;
    v8f acc[2] = {};

    for (int kk = 0; kk < GD; kk += 4) {
        int ka = kk + (half << 1);
        v2f a;
        a.x = arow[ka];
        a.y = arow[ka + 1];
        const float* w0 = W1 + ka * GH + l16;
        v2f b0, b1v;
        b0.x  = w0[0];        // W1[ka][l16]
        b0.y  = w0[GH];       // W1[ka+1][l16]
        b1v.x = w0[16];       // W1[ka][16+l16]
        b1v.y = w0[GH + 16];  // W1[ka+1][16+l16]
        acc[0] = __builtin_amdgcn_wmma_f32_16x16x4_f32(false, a, false, b0,
                                                       (short)0, acc[0], false, false);
        acc[1] = __builtin_amdgcn_wmma_f32_16x16x4_f32(false, a, false, b1v,
                                                       (short)0, acc[1], false, false);
    }

    int row0 = wave * 16 + (half << 3);
#pragma unroll
    for (int t = 0; t < 2; ++t) {
        int col = t * 16 + l16;
        float bias = b1[col];
        float s = 0.0f, ss = 0.0f;
#pragma unroll
        for (int j = 0; j < 8; ++j) {
            float v = acc[t][j] + bias;
            v = v > 0.0f ? v : 0.0f;                 // ReLU
            Hout[(size_t)(row0 + j) * GH + col] = v;
            s += v;
            ss += v * v;
        }
        atomicAdd(&stats[col], s);
        atomicAdd(&stats[GH + col], ss);
    }
}

// ---------------- BN finalize: a = g*rsqrt(var+eps), c = be - m*a ----------------

__global__ void finalize_stats(const float* __restrict__ stats, const float* __restrict__ g,
                               const float* __restrict__ be, float* __restrict__ ac,
                               int C, float invN) {
    int k = threadIdx.x;
    if (k >= C) return;
    float m = stats[k] * invN;
    float v = stats[C + k] * invN - m * m;
    float inv = rsqrtf(v + BN_EPS);
    float a = g[k] * inv;
    ac[k] = a;
    ac[C + k] = be[k] - m * a;
}

// ---------------- fold BN1 into W2: W2' = diag(a1) W2, b2' = b2 + c1 @ W2 ----------

__global__ void fold_w2(const float* __restrict__ W2, const float* __restrict__ b2,
                        const float* __restrict__ a1c1, float* __restrict__ W2p,
                        float* __restrict__ b2p) {
    int j = threadIdx.x;  // 0..127
    float bsum = b2[j];
    for (int k = 0; k < GH; ++k) {
        float a = a1c1[k];
        float c = a1c1[GH + k];
        float w = W2[k * GD + j];
        W2p[k * GD + j] = a * w;
        bsum += c * w;
    }
    b2p[j] = bsum;
}

// ---------------- GEMM2: U = Hn @ W2' + b2', + BN stats ----------------
// one wave per 16-row tile; K=32; 16x128 output per wave (8 accumulators).

__global__ void __launch_bounds__(256)
gemm2_stats(const float* __restrict__ Hin, const float* __restrict__ W2p,
            const float* __restrict__ b2p, float* __restrict__ U,
            float* __restrict__ stats, int ntiles) {
    int wave = (blockIdx.x << 3) + (threadIdx.x >> 5);
    if (wave >= ntiles) return;
    int lane = threadIdx.x & 31;
    int half = lane >> 4;
    int l16  = lane & 15;

    const float* arow = Hin + (size_t)(wave * 16 + l16) * GH;
    v8f acc[8] = {};

    for (int kk = 0; kk < GH; kk += 4) {
        int ka = kk + (half << 1);
        v2f a;
        a.x = arow[ka];
        a.y = arow[ka + 1];
        const float* w0 = W2p + ka * GD + l16;
#pragma unroll
        for (int nt = 0; nt < 8; ++nt) {
            v2f b;
            b.x = w0[nt * 16];
            b.y = w0[GD + nt * 16];
            acc[nt] = __builtin_amdgcn_wmma_f32_16x16x4_f32(false, a, false, b,
                                                            (short)0, acc[nt], false, false);
        }
    }

    int row0 = wave * 16 + (half << 3);
#pragma unroll
    for (int nt = 0; nt < 8; ++nt) {
        int col = nt * 16 + l16;
        float bias = b2p[col];
        float s = 0.0f, ss = 0.0f;
#pragma unroll
        for (int j = 0; j < 8; ++j) {
            float v = acc[nt][j] + bias;
            U[(size_t)(row0 + j) * GD + col] = v;
            s += v;
            ss += v * v;
        }
        atomicAdd(&stats[col], s);
        atomicAdd(&stats[GD + col], ss);
    }
}

// ---------------- apply BN2 + LeakyReLU; optionally seed next agg buffer ----------

__global__ void __launch_bounds__(256)
apply_bn_leaky(const float* __restrict__ U, const float* __restrict__ ac,
               float* __restrict__ Y, float* __restrict__ AggInit, int n) {
    int i = blockIdx.x * blockDim.x + threadIdx.x;
    if (i >= n) return;
    int col = i & (GD - 1);
    float v = U[i] * ac[col] + ac[GD + col];
    float y = v > 0.0f ? v : NEG_SLOPE * v;
    Y[i] = y;
    if (AggInit) AggInit[i] = y;
}

// ---------------------------------------------------------------------------

extern "C" void kernel_launch(void* const* d_in, const int* in_sizes, int n_in,
                              void* d_out, int out_size, void* d_ws, size_t ws_size,
                              hipStream_t stream) {
    const float* x   = (const float*)d_in[0];
    const int*   src = (const int*)d_in[1];
    const int*   dst = (const int*)d_in[2];

    // workspace layout (floats)
    float* bufA   = (float*)d_ws;                 // [N,D] agg / U scratch
    float* bufC   = bufA + (size_t)GN * GD;       // [N,D] layer-1 output y1
    float* bufB   = bufC + (size_t)GN * GD;       // [N,H] hidden
    float* stats1 = bufB + (size_t)GN * GH;       // 2*H
    float* stats2 = stats1 + 2 * GH;              // 2*D
    float* a1c1   = stats2 + 2 * GD;              // 2*H
    float* a2c2   = a1c1 + 2 * GH;                // 2*D
    float* W2p    = a2c2 + 2 * GD;                // H*D
    float* b2p    = W2p + GH * GD;                // D

    const int ntiles   = GN / 16;                 // 6250, exact
    const int nElemND  = GN * GD;                 // 12.8M
    const int nScatter = GE * 32;                 // 51.2M threads
    dim3 gGemm((ntiles + 7) / 8);
    dim3 gND((nElemND + 255) / 256);
    dim3 gSc((nScatter + 255) / 256);
    const float invN = 1.0f / (float)GN;

    for (int l = 0; l < 2; ++l) {
        const float* W1l  = (const float*)d_in[3 + 8 * l];
        const float* b1l  = (const float*)d_in[4 + 8 * l];
        const float* g1l  = (const float*)d_in[5 + 8 * l];
        const float* be1l = (const float*)d_in[6 + 8 * l];
        const float* W2l  = (const float*)d_in[7 + 8 * l];
        const float* b2l  = (const float*)d_in[8 + 8 * l];
        const float* g2l  = (const float*)d_in[9 + 8 * l];
        const float* be2l = (const float*)d_in[10 + 8 * l];

        const float* xin = (l == 0) ? x : bufC;   // gather source for this layer

        if (l == 0)
            copy_f32<<<gND, 256, 0, stream>>>(bufA, x, nElemND);  // self term
        // (for l==1, apply_bn_leaky already seeded bufA with y1)

        zero_f32<<<1, 2 * GH + 2 * GD, 0, stream>>>(stats1, 2 * GH + 2 * GD);

        scatter_edges<<<gSc, 256, 0, stream>>>(bufA, xin, src, dst, nScatter);

        gemm1_relu_stats<<<gGemm, 256, 0, stream>>>(bufA, W1l, b1l, bufB, stats1, ntiles);
        finalize_stats<<<1, GH, 0, stream>>>(stats1, g1l, be1l, a1c1, GH, invN);
        fold_w2<<<1, GD, 0, stream>>>(W2l, b2l, a1c1, W2p, b2p);

        gemm2_stats<<<gGemm, 256, 0, stream>>>(bufB, W2p, b2p, bufA, stats2, ntiles);
        finalize_stats<<<1, GD, 0, stream>>>(stats2, g2l, be2l, a2c2, GD, invN);

        float* yout    = (l == 0) ? bufC : (float*)d_out;
        float* agginit = (l == 0) ? bufA : nullptr;
        apply_bn_leaky<<<gND, 256, 0, stream>>>(bufA, a2c2, yout, agginit, nElemND);
    }
}